// fdm_expand_block_new_30313879175594
// MI455X (gfx1250) — compile-verified
//
#include <hip/hip_runtime.h>

// Sizes fixed by the reference: b=4, f=64, h=w=64
#define Bn 4
#define Fc 64
#define Hh 64
#define Ww 64
#define HW 4096            // 64*64
#define NEG_HUGE 3.4028235e38f
#define ROWSLICE 1024      // rows per k_simmax workgroup (grid.z = 4 slices)
#define USE_ASYNC 1        // gfx1250 GLOBAL_LOAD_ASYNC_TO_LDS double-buffering

typedef float v2f __attribute__((ext_vector_type(2)));
typedef float v8f __attribute__((ext_vector_type(8)));

// Order-preserving float<->uint key (works for negative values, unlike raw bits)
__device__ __forceinline__ unsigned fkey(float f) {
    unsigned b = __float_as_uint(f);
    return (b & 0x80000000u) ? ~b : (b | 0x80000000u);
}
__device__ __forceinline__ float funkey(unsigned k) {
    return __uint_as_float((k & 0x80000000u) ? (k & 0x7FFFFFFFu) : ~k);
}
#define KEY_NEG_MAX 0x00800000u   // fkey(-FLT_MAX)

// ---------------------------------------------------------------------------
// misc layout (floats): [0..3]=bmax per batch, [4..7]=gn sum, [8..11]=gn sumsq
// smkey: 16384 u32 partial-max keys for sm
// ---------------------------------------------------------------------------
__global__ __launch_bounds__(256)
void k_initall(unsigned* __restrict__ smkey, float* __restrict__ misc) {
    const int gid = blockIdx.x * 256 + threadIdx.x;
    if (gid < Bn * HW) smkey[gid] = KEY_NEG_MAX;
    if (gid < 12) misc[gid] = 0.0f;
}

// 2x2 max pool of fdm [4,1,128,128] -> fdm_p [4,4096]; per-batch max via atomic
__global__ __launch_bounds__(256)
void k_pool(const float* __restrict__ fdm, float* __restrict__ fdm_p,
            float* __restrict__ misc) {
    const int b = blockIdx.y;
    const int p = blockIdx.x * 256 + threadIdx.x;      // 16 blocks * 256 = 4096
    const int py = p >> 6, px = p & 63;
    const float* f = fdm + b * (128 * 128);
    const int r = py * 2, c = px * 2;
    float m = fmaxf(fmaxf(f[r * 128 + c],       f[r * 128 + c + 1]),
                    fmaxf(f[(r + 1) * 128 + c], f[(r + 1) * 128 + c + 1]));
    fdm_p[b * HW + p] = m;

    __shared__ float red[256];
    red[threadIdx.x] = m;
    __syncthreads();
    for (int s = 128; s > 0; s >>= 1) {
        if (threadIdx.x < s) red[threadIdx.x] = fmaxf(red[threadIdx.x], red[threadIdx.x + s]);
        __syncthreads();
    }
    // fdm values are >= 0, so int-compare on float bits is monotonic
    if (threadIdx.x == 0) atomicMax((int*)&misc[b], __float_as_int(red[0]));
}

// mask[b,p] = fdm_p >= min(0.5, bmax/3)
__global__ __launch_bounds__(256)
void k_mask(const float* __restrict__ fdm_p, const float* __restrict__ misc,
            float* __restrict__ mask) {
    const int b = blockIdx.y;
    const int p = blockIdx.x * 256 + threadIdx.x;
    const float thre = fminf(0.5f, misc[b] * (1.0f / 3.0f));
    mask[b * HW + p] = (fdm_p[b * HW + p] >= thre) ? 1.0f : 0.0f;
}

// xn[b][p][c] = x[b][c][p] / max(||x[b][:,p]||, 1e-8)
__global__ __launch_bounds__(256)
void k_norm(const float* __restrict__ x, float* __restrict__ xn) {
    const int gid = blockIdx.x * 256 + threadIdx.x;    // 0..16383
    const int b = gid >> 12, p = gid & 4095;
    const float* xb = x + (size_t)b * Fc * HW;
    float s = 0.0f;
#pragma unroll
    for (int c = 0; c < Fc; ++c) { float v = xb[c * HW + p]; s += v * v; }
    const float inv = 1.0f / fmaxf(sqrtf(s), 1e-8f);
    float* o = xn + ((size_t)b * HW + p) * Fc;
#pragma unroll
    for (int c = 0; c < Fc; ++c) o[c] = xb[c * HW + p] * inv;
}

// ---------------------------------------------------------------------------
// sim = Xn * Xn^T  (per batch, 4096x4096, K=64) via V_WMMA_F32_16X16X4_F32,
// fused masked row-max. Grid (16, 4, 4): x = 256-column tile, y = batch,
// z = 1024-row slice. Partial column maxima combined with keyed atomicMax.
// B fragments (32 cols x 16 k-steps) live in registers for the whole slice;
// the 16-row A panel is double-buffered in LDS via async global->LDS copies.
// ---------------------------------------------------------------------------
__global__ __launch_bounds__(256)
void k_simmax(const float* __restrict__ xn, const float* __restrict__ mask,
              unsigned* __restrict__ smkey) {
    const int b = blockIdx.y;
    const int rowbase_wg = blockIdx.z * ROWSLICE;
    const int colbase_wg = blockIdx.x * 256;
    const int tid  = threadIdx.x;
    const int wave = tid >> 5;
    const int lane = tid & 31;
    const int lmod = lane & 15;     // N (for B/C) or M (for A)
    const int half = lane >> 4;     // 0/1 -> K offset / M offset
    const int k0 = half * 2;

    const float* xnb = xn + (size_t)b * HW * Fc;

    // B fragments: 2 N-tiles x 16 k-steps (K=64 total, 4 per WMMA)
    v2f bfrag[2][16];
#pragma unroll
    for (int t = 0; t < 2; ++t) {
        const int q = colbase_wg + wave * 32 + t * 16 + lmod;
        const float* col = xnb + (size_t)q * Fc;
#pragma unroll
        for (int kk = 0; kk < 16; ++kk) {
            bfrag[t][kk].x = col[kk * 4 + k0];
            bfrag[t][kk].y = col[kk * 4 + k0 + 1];
        }
    }

    __shared__ float ldsA[2][16 * 64];     // double-buffered A panel (2 x 4 KB)
    __shared__ float ldsMask[ROWSLICE];    // row mask for this slice (4 KB)

#pragma unroll
    for (int i = 0; i < ROWSLICE / 256; ++i) {
        const int idx = tid + i * 256;
        ldsMask[idx] = mask[b * HW + rowbase_wg + idx];
    }

#if USE_ASYNC
    {   // prefetch chunk 0: 256 threads x 16 B = 4 KB panel
        unsigned dst = (unsigned)(size_t)&ldsA[0][tid * 4];
        const float* src = xnb + (size_t)(rowbase_wg + (tid >> 4)) * Fc + (tid & 15) * 4;
        asm volatile("global_load_async_to_lds_b128 %0, %1, off" :: "v"(dst), "v"(src) : "memory");
    }
#endif

    float cmax0 = -NEG_HUGE, cmax1 = -NEG_HUGE;

    for (int ch = 0; ch < ROWSLICE / 16; ++ch) {
        const int pbase = rowbase_wg + ch * 16;
#if USE_ASYNC
        if (ch < ROWSLICE / 16 - 1) {      // prefetch next panel into other buffer
            unsigned dst = (unsigned)(size_t)&ldsA[(ch + 1) & 1][tid * 4];
            const float* src = xnb + (size_t)(pbase + 16 + (tid >> 4)) * Fc + (tid & 15) * 4;
            asm volatile("global_load_async_to_lds_b128 %0, %1, off" :: "v"(dst), "v"(src) : "memory");
            asm volatile("s_wait_asynccnt 0x1" ::: "memory");  // chunk ch done (in-order)
        } else {
            asm volatile("s_wait_asynccnt 0x0" ::: "memory");
        }
        __syncthreads();                   // panel visible to all waves
        const float* A = ldsA[ch & 1];
#else
#pragma unroll
        for (int i = 0; i < 4; ++i) {
            const int idx = tid + i * 256;
            ldsA[0][idx] = xnb[(size_t)(pbase + (idx >> 6)) * Fc + (idx & 63)];
        }
        __syncthreads();
        const float* A = ldsA[0];
#endif

        v8f c0 = {}; v8f c1 = {};
#pragma unroll
        for (int kk = 0; kk < 16; ++kk) {
            v2f a;
            a.x = A[lmod * 64 + kk * 4 + k0];
            a.y = A[lmod * 64 + kk * 4 + k0 + 1];
            c0 = __builtin_amdgcn_wmma_f32_16x16x4_f32(false, a, false, bfrag[0][kk],
                                                       (short)0, c0, false, false);
            c1 = __builtin_amdgcn_wmma_f32_16x16x4_f32(false, a, false, bfrag[1][kk],
                                                       (short)0, c1, false, false);
        }

        // masked max over this 16-row chunk (C: VGPR r -> M = r + 8*half)
#pragma unroll
        for (int r = 0; r < 8; ++r) {
            const float m = ldsMask[ch * 16 + r + half * 8];
            cmax0 = fmaxf(cmax0, (m > 0.5f) ? c0[r] : -NEG_HUGE);
            cmax1 = fmaxf(cmax1, (m > 0.5f) ? c1[r] : -NEG_HUGE);
        }
        __syncthreads();                   // reads done before buffer is overwritten
    }

    // lanes L and L+16 cover the same column N with different M halves
    cmax0 = fmaxf(cmax0, __shfl_xor(cmax0, 16, 32));
    cmax1 = fmaxf(cmax1, __shfl_xor(cmax1, 16, 32));
    if (half == 0) {
        const int q0 = colbase_wg + wave * 32 + lmod;
        atomicMax(&smkey[b * HW + q0],      fkey(cmax0));
        atomicMax(&smkey[b * HW + q0 + 16], fkey(cmax1));
    }
}

// sm = decode(key)/2 + 0.5
__global__ __launch_bounds__(256)
void k_smfinal(const unsigned* __restrict__ smkey, float* __restrict__ sm_out) {
    const int gid = blockIdx.x * 256 + threadIdx.x;    // 0..16383
    sm_out[gid] = funkey(smkey[gid]) * 0.5f + 0.5f;
}

// ---------------------------------------------------------------------------
// Conv2d(128->64, 3x3, pad 1) on cat = [sm*x ; x] as implicit GEMM (WMMA),
// + bias, + per-batch sum/sumsq for GroupNorm. 128 threads (4 waves), each
// wave one 16oc x 16pos C tile; 16 positions per block -> LDS ~61 KB.
// ---------------------------------------------------------------------------
__global__ __launch_bounds__(128)
void k_conv(const float* __restrict__ x, const float* __restrict__ sm,
            const float* __restrict__ Wf, const float* __restrict__ bias,
            float* __restrict__ y, float* __restrict__ misc) {
    const int b = blockIdx.y;
    const int tile = blockIdx.x;            // 0..255
    const int row = tile >> 2;
    const int colbase = (tile & 3) * 16;
    const int tid  = threadIdx.x;
    const int wave = tid >> 5;              // 0..3
    const int lane = tid & 31;
    const int lmod = lane & 15;
    const int half = lane >> 4;
    const int k0 = half * 2;
    const int ocbase = wave * 16;

    __shared__ float ldsCat[3 * 128 * 18];  // rows row-1..row+1, 128 ch, 18 cols (halo+pad)
    __shared__ float ldsW[64 * 128];        // per-tap weight slice [oc][ic]
    __shared__ float red[128], redq[128];

    const float* xb  = x  + (size_t)b * Fc * HW;
    const float* smb = sm + b * HW;

    for (int idx = tid; idx < 3 * 128 * 18; idx += 128) {
        const int rr  = idx / (128 * 18);
        const int rem = idx - rr * (128 * 18);
        const int ic  = rem / 18;
        const int cc  = rem - ic * 18;
        const int irow = row - 1 + rr;
        const int icol = colbase - 1 + cc;
        float v = 0.0f;
        if (irow >= 0 && irow < Hh && icol >= 0 && icol < Ww) {
            const int pos = irow * Ww + icol;
            const float xv = xb[(ic & 63) * HW + pos];
            v = (ic < 64) ? smb[pos] * xv : xv;
        }
        ldsCat[idx] = v;
    }

    v8f acc = {};
    for (int tap = 0; tap < 9; ++tap) {
        const int kh = tap / 3, kw = tap - kh * 3;
        __syncthreads();                               // guard ldsW restage / ldsCat ready
        for (int idx = tid; idx < 64 * 128; idx += 128) {
            const int oc = idx >> 7, ic = idx & 127;
            ldsW[idx] = Wf[((oc * 128 + ic) * 3 + kh) * 3 + kw];
        }
        __syncthreads();

        const float* aLds = &ldsW[(ocbase + lmod) * 128];
#pragma unroll 4
        for (int ic0 = 0; ic0 < 128; ic0 += 4) {
            v2f a;  a.x = aLds[ic0 + k0];  a.y = aLds[ic0 + k0 + 1];
            const float* bp = &ldsCat[(kh * 128 + ic0 + k0) * 18 + lmod + kw];
            v2f bf; bf.x = bp[0]; bf.y = bp[18];
            acc = __builtin_amdgcn_wmma_f32_16x16x4_f32(false, a, false, bf,
                                                        (short)0, acc, false, false);
        }
    }

    float lsum = 0.0f, lsq = 0.0f;
    float* yb = y + (size_t)b * Fc * HW;
#pragma unroll
    for (int r = 0; r < 8; ++r) {
        const int oc = ocbase + r + half * 8;
        const float v = acc[r] + bias[oc];
        yb[oc * HW + row * Ww + colbase + lmod] = v;
        lsum += v; lsq += v * v;
    }
    red[tid] = lsum; redq[tid] = lsq;
    __syncthreads();
    for (int s = 64; s > 0; s >>= 1) {
        if (tid < s) { red[tid] += red[tid + s]; redq[tid] += redq[tid + s]; }
        __syncthreads();
    }
    if (tid == 0) {
        atomicAdd(&misc[4 + b], red[0]);
        atomicAdd(&misc[8 + b], redq[0]);
    }
}

// GroupNorm(1 group, eps=1e-5) + affine + ReLU
__global__ __launch_bounds__(256)
void k_gn(const float* __restrict__ y, const float* __restrict__ misc,
          const float* __restrict__ gamma, const float* __restrict__ beta,
          float* __restrict__ out) {
    const int gid = blockIdx.x * 256 + threadIdx.x;    // 0 .. 4*64*4096
    const int b = gid >> 18;
    const int c = (gid & 262143) >> 12;
    const float N = 262144.0f;
    const float mean = misc[4 + b] / N;
    const float var  = misc[8 + b] / N - mean * mean;
    const float rstd = rsqrtf(var + 1e-5f);
    const float v = (y[gid] - mean) * rstd * gamma[c] + beta[c];
    out[gid] = v > 0.0f ? v : 0.0f;
}

// ---------------------------------------------------------------------------
extern "C" void kernel_launch(void* const* d_in, const int* in_sizes, int n_in,
                              void* d_out, int out_size, void* d_ws, size_t ws_size,
                              hipStream_t stream) {
    (void)in_sizes; (void)n_in; (void)out_size; (void)ws_size;
    const float* fdm   = (const float*)d_in[0];   // [4,1,128,128]
    const float* x     = (const float*)d_in[1];   // [4,64,64,64]
    const float* Wf    = (const float*)d_in[2];   // [64,128,3,3]
    const float* bfuse = (const float*)d_in[3];   // [64]
    const float* gamma = (const float*)d_in[4];   // [64]
    const float* beta  = (const float*)d_in[5];   // [64]

    float*    xn    = (float*)d_ws;               // 1,048,576 f
    float*    y     = xn + 1048576;               // 1,048,576 f
    float*    fdm_p = y + 1048576;                // 16,384 f
    float*    mask  = fdm_p + 16384;              // 16,384 f
    unsigned* smkey = (unsigned*)(mask + 16384);  // 16,384 u32
    float*    misc  = (float*)(smkey + 16384);    // 12 f

    float* x0_out = (float*)d_out;                // [4,64,64,64]
    float* sm_out = x0_out + 1048576;             // [4,1,64,64]

    k_initall<<<64, 256, 0, stream>>>(smkey, misc);
    k_pool   <<<dim3(16, Bn), 256, 0, stream>>>(fdm, fdm_p, misc);
    k_mask   <<<dim3(16, Bn), 256, 0, stream>>>(fdm_p, misc, mask);
    k_norm   <<<64, 256, 0, stream>>>(x, xn);
    k_simmax <<<dim3(16, Bn, 4), 256, 0, stream>>>(xn, mask, smkey);
    k_smfinal<<<64, 256, 0, stream>>>(smkey, sm_out);
    k_conv   <<<dim3(256, Bn), 128, 0, stream>>>(x, sm_out, Wf, bfuse, y, misc);
    k_gn     <<<4096, 256, 0, stream>>>(y, misc, gamma, beta, x0_out);
}